// X_plus_Layer_47339129536788
// MI455X (gfx1250) — compile-verified
//
#include <hip/hip_runtime.h>
#include <cstdint>

typedef float v4f __attribute__((ext_vector_type(4)));

#define ROW_LEN 75u          // 25 * 3 contiguous elements share one x value
#define BLK     256          // 8 wave32 waves
#define CHUNK   (BLK * 4u)   // floats per block = 1024 -> spans <= 15 rows
#define NSLOT   16           // staged x values per block (>= 15)

#if defined(__HIP_DEVICE_COMPILE__) && __has_builtin(__builtin_amdgcn_global_load_async_to_lds_b32)
#define HAVE_ASYNC_LDS 1
#endif

__global__ __launch_bounds__(BLK) void xscale_bcast_kernel(
    const float* __restrict__ x,      // [B*N]
    const v4f*   __restrict__ A4,     // [B*N*75/4] viewed as float4
    const float* __restrict__ alphap, // scalar
    const float* __restrict__ betap,  // scalar
    v4f*         __restrict__ out4,   // same shape as A4
    unsigned nrows,                   // B*N
    unsigned total_f4)                // total float4 count
{
    __shared__ float xs[NSLOT];

    const unsigned tid       = threadIdx.x;
    const unsigned blockBase = blockIdx.x * CHUNK;      // first float index of block
    const unsigned rFirst    = blockBase / ROW_LEN;     // first x row touched

    // ---- stage the x rows this block needs into LDS (CDNA5 async path) ----
    if (tid < NSLOT) {
        unsigned r = rFirst + tid;
        if (r >= nrows) r = nrows - 1u;                 // clamp (slots 15.. unused anyway)
#ifdef HAVE_ASYNC_LDS
        __builtin_amdgcn_global_load_async_to_lds_b32(
            (__attribute__((address_space(1))) int*)(x + r),
            (__attribute__((address_space(3))) int*)(xs + tid),
            /*offset=*/0, /*cpol=*/0);
#else
        xs[tid] = x[r];
#endif
    }
#ifdef HAVE_ASYNC_LDS
#if __has_builtin(__builtin_amdgcn_s_wait_asynccnt)
    __builtin_amdgcn_s_wait_asynccnt(0);
#else
    asm volatile("s_wait_asynccnt 0" ::: "memory");
#endif
#endif
    __syncthreads();

    const float alpha = *alphap;   // uniform -> scalar load, L2 cached
    const float beta  = *betap;

    const unsigned i4 = blockBase / 4u + tid;           // float4 index
    if (i4 >= total_f4) return;

    const unsigned base = blockBase + tid * 4u;         // float index (< 2^31)
    const unsigned r0   = base / ROW_LEN;               // magic-mul division
    const unsigned rem  = base - r0 * ROW_LEN;          // 0..74
    const unsigned s0   = r0 - rFirst;                  // LDS slot, <= 14

    const float x0 = xs[s0];
    const float x1 = xs[s0 + (rem >= 72u ? 1u : 0u)];   // row straddle (rem 72..74)

    // Streamed, single-use: non-temporal 128-bit load.
    v4f a = __builtin_nontemporal_load(A4 + i4);

    // out = x * (alpha*A + beta)
    const float t0 = fmaf(alpha, a.x, beta);
    const float t1 = fmaf(alpha, a.y, beta);
    const float t2 = fmaf(alpha, a.z, beta);
    const float t3 = fmaf(alpha, a.w, beta);

    v4f o;
    o.x = x0 * t0;                         // rem+0 < 75 always
    o.y = (rem >= 74u ? x1 : x0) * t1;
    o.z = (rem >= 73u ? x1 : x0) * t2;
    o.w = (rem >= 72u ? x1 : x0) * t3;

    // Write-once output: non-temporal 128-bit store.
    __builtin_nontemporal_store(o, out4 + i4);
}

extern "C" void kernel_launch(void* const* d_in, const int* in_sizes, int n_in,
                              void* d_out, int out_size, void* d_ws, size_t ws_size,
                              hipStream_t stream) {
    const float* x      = (const float*)d_in[0];   // [B,N,1]
    const float* A      = (const float*)d_in[1];   // [B,N,25,3]
    const float* alphap = (const float*)d_in[2];   // scalar
    const float* betap  = (const float*)d_in[3];   // scalar

    const unsigned nrows    = (unsigned)in_sizes[0];         // B*N = 1,228,800
    const unsigned total    = (unsigned)out_size;            // 92,160,000 floats
    const unsigned total_f4 = total / 4u;                    // 23,040,000 float4s
    const unsigned nblocks  = (total + CHUNK - 1u) / CHUNK;  // 90,000 (exact, no tail)

    xscale_bcast_kernel<<<dim3(nblocks), dim3(BLK), 0, stream>>>(
        x, (const v4f*)A, alphap, betap, (v4f*)d_out, nrows, total_f4);
}